// QuonvLayer_83623013253698
// MI455X (gfx1250) — compile-verified
//
#include <hip/hip_runtime.h>
#include <hip/hip_bf16.h>

typedef __attribute__((ext_vector_type(2))) float v2f;
typedef __attribute__((ext_vector_type(8))) float v8f;

#define N_PATCHES_TOTAL (16 * 191 * 191)     // 583696
#define N_TILES         (N_PATCHES_TOTAL/16) // 36481 (exact)
#define PATCH_PER_IMG   (191 * 191)          // 36481
#define KDIM            84                   // 81 trig-basis components padded to 4*21

// ---------------------------------------------------------------------------
// cf(t,a,b): coefficient of basis function t in the per-wire outer product
//   cos(x/2),sin(x/2) outer factors:  c^2=(1+C)/2, s^2=(1-C)/2, cs=S/2
//   basis: t=0 -> 1, t=1 -> C=cos(x), t=2 -> S=sin(x)
// ---------------------------------------------------------------------------
__device__ __forceinline__ float qc_cf(int t, int a, int b) {
    if (t == 0) return (a == b) ? 0.5f : 0.0f;
    if (t == 1) return (a == b) ? (a ? -0.5f : 0.5f) : 0.0f;
    return (a != b) ? 0.5f : 0.0f;
}

// ---------------------------------------------------------------------------
// Setup kernel (run once per launch, 1 block of 512 threads):
//  1) threads 0..15 simulate column |i> through 2x(RX layer + CNOT ring)
//  2) A_w = Re(V^dag Z_w V)  (real symmetric 16x16, 4 of them)
//  3) project onto 3^4 trig-product basis -> M84[84][16] in d_ws
//     (only columns 0..3 / rows 0..80 nonzero; rest written as 0)
// ---------------------------------------------------------------------------
__global__ void quonv_setup_kernel(const float* __restrict__ wts,  // [2][4]
                                   float* __restrict__ M84)        // [84][16]
{
    __shared__ float Vre[16][16];   // [state m][input column i]
    __shared__ float Vim[16][16];
    __shared__ float Amat[4][16][16];

    const int tid = threadIdx.x;

    // ---- phase 1: simulate columns (wire w <-> bit (3-w), C-order flatten)
    if (tid < 16) {
        #pragma unroll
        for (int m = 0; m < 16; ++m) {
            Vre[m][tid] = (m == tid) ? 1.0f : 0.0f;
            Vim[m][tid] = 0.0f;
        }
        #pragma unroll
        for (int l = 0; l < 2; ++l) {
            // RX(theta_{l,w}) on each wire
            #pragma unroll
            for (int w = 0; w < 4; ++w) {
                const float th = wts[l * 4 + w];
                const float c = cosf(0.5f * th);
                const float s = sinf(0.5f * th);
                const int msk = 1 << (3 - w);
                #pragma unroll
                for (int k = 0; k < 16; ++k) {
                    if (!(k & msk)) {
                        const int k1 = k | msk;
                        float ar = Vre[k][tid],  ai = Vim[k][tid];
                        float br = Vre[k1][tid], bi = Vim[k1][tid];
                        // RX = [[c, -i s],[-i s, c]]
                        Vre[k][tid]  = c * ar + s * bi;
                        Vim[k][tid]  = c * ai - s * br;
                        Vre[k1][tid] = c * br + s * ai;
                        Vim[k1][tid] = c * bi - s * ar;
                    }
                }
            }
            // CNOT ring (w -> (w+1)%4), sequential as in reference
            #pragma unroll
            for (int w = 0; w < 4; ++w) {
                const int mc = 1 << (3 - w);
                const int mt = 1 << (3 - ((w + 1) & 3));
                #pragma unroll
                for (int k = 0; k < 16; ++k) {
                    if ((k & mc) && !(k & mt)) {
                        const int k1 = k | mt;
                        float tr = Vre[k][tid]; Vre[k][tid] = Vre[k1][tid]; Vre[k1][tid] = tr;
                        float ti = Vim[k][tid]; Vim[k][tid] = Vim[k1][tid]; Vim[k1][tid] = ti;
                    }
                }
            }
        }
    }
    __syncthreads();

    // ---- phase 2: A_w[i][k] = sum_m sign_w(m) * Re(conj(V[m][i]) V[m][k])
    for (int idx = tid; idx < 4 * 16 * 16; idx += blockDim.x) {
        const int w = idx >> 8, i = (idx >> 4) & 15, k = idx & 15;
        const int mw = 1 << (3 - w);
        float s = 0.0f;
        #pragma unroll
        for (int m = 0; m < 16; ++m) {
            const float sgn = (m & mw) ? -1.0f : 1.0f;
            s += sgn * (Vre[m][i] * Vre[m][k] + Vim[m][i] * Vim[m][k]);
        }
        Amat[w][i][k] = s;
    }
    __syncthreads();

    // ---- phase 3: M84[j][col] = sum_{i,k} A_col[i][k] * prod_w cf(j_w, i_w, k_w)
    for (int idx = tid; idx < KDIM * 16; idx += blockDim.x) {
        const int j = idx >> 4, col = idx & 15;
        float val = 0.0f;
        if (j < 81 && col < 4) {
            const int j3 = j % 3, j2 = (j / 3) % 3, j1 = (j / 9) % 3, j0 = j / 27;
            float s = 0.0f;
            for (int i = 0; i < 16; ++i) {
                for (int k = 0; k < 16; ++k) {
                    const float p = qc_cf(j0, (i >> 3) & 1, (k >> 3) & 1) *
                                    qc_cf(j1, (i >> 2) & 1, (k >> 2) & 1) *
                                    qc_cf(j2, (i >> 1) & 1, (k >> 1) & 1) *
                                    qc_cf(j3,  i       & 1,  k       & 1);
                    if (p != 0.0f) s += Amat[col][i][k] * p;
                }
            }
            val = s;
        }
        M84[idx] = val;
    }
}

// ---------------------------------------------------------------------------
// Main kernel: one wave32 handles one 16-patch tile.
//   A-operand (16x4 f32): lane = 16*g + p  holds  (M=p, K=2g+v) in vgpr v
//   B-operand (4x16 f32): lane = 16*g + p  holds  (K=2g+v, N=p) in vgpr v
//   D (16x16 f32): vgpr r, lane (g,p) holds (M=r+8g, N=p)
//   out[patch, w] sits at D[M=patch-in-tile][N=w], w<4.
// ---------------------------------------------------------------------------
__global__ void __launch_bounds__(256)
quonv_main_kernel(const float* __restrict__ img,   // [16][192][192]
                  const float* __restrict__ M84,   // [84][16]
                  float* __restrict__ out)         // [583696][4]
{
    const int lane = threadIdx.x & 31;
    const int wave = (blockIdx.x * blockDim.x + threadIdx.x) >> 5;
    const int g = lane >> 4;       // lane group (selects K pair)
    const int p = lane & 15;       // M (patch row) for A, N (column) for B/D

    if (wave >= N_TILES) return;   // wave-uniform: EXEC stays all-1s below

    // --- load B fragments (shared coefficient matrix, hot in L2)
    v2f bfrag[21];
    #pragma unroll
    for (int c = 0; c < 21; ++c) {
        const int k0 = 4 * c + 2 * g;
        bfrag[c][0] = M84[ k0      * 16 + p];
        bfrag[c][1] = M84[(k0 + 1) * 16 + p];
    }

    // --- per-patch trig features (lanes p and p+16 redundantly compute patch p)
    const int P0 = wave * 16;
    const int P  = P0 + p;
    const int b   = P / PATCH_PER_IMG;
    const int rem = P - b * PATCH_PER_IMG;
    const int oy  = rem / 191;
    const int ox  = rem - oy * 191;
    const float* src = img + (b * 192 + oy) * 192 + ox;
    const float x0 = src[0], x1 = src[1], x2 = src[192], x3 = src[193];

    float S0, C0, S1, C1, S2, C2, S3, C3;
    __sincosf(x0, &S0, &C0);
    __sincosf(x1, &S1, &C1);
    __sincosf(x2, &S2, &C2);
    __sincosf(x3, &S3, &C3);

    // tensor-product halves of v81 = (1,C0,S0)x(1,C1,S1)x(1,C2,S2)x(1,C3,S3)
    const float t01[9] = {1.0f, C1, S1, C0, C0*C1, C0*S1, S0, S0*C1, S0*S1};
    const float t23[9] = {1.0f, C3, S3, C2, C2*C3, C2*S3, S2, S2*C3, S2*S3};

#define QV81(J) (((J) < 81) ? (t01[(J) / 9] * t23[(J) % 9]) : 0.0f)

    // A fragments: constant indices inside each g-branch (reconverges before WMMA)
    v2f frag[21];
    if (g == 0) {
        #pragma unroll
        for (int c = 0; c < 21; ++c) {
            frag[c][0] = QV81(4 * c + 0);
            frag[c][1] = QV81(4 * c + 1);
        }
    } else {
        #pragma unroll
        for (int c = 0; c < 21; ++c) {
            frag[c][0] = QV81(4 * c + 2);
            frag[c][1] = QV81(4 * c + 3);
        }
    }
#undef QV81

    // --- full-precision f32 WMMA accumulate chain over K = 84
    v8f acc = {0.0f, 0.0f, 0.0f, 0.0f, 0.0f, 0.0f, 0.0f, 0.0f};
    #pragma unroll
    for (int c = 0; c < 21; ++c) {
        acc = __builtin_amdgcn_wmma_f32_16x16x4_f32(
                  /*neg_a=*/false, frag[c],
                  /*neg_b=*/false, bfrag[c],
                  /*c_mod=*/(short)0, acc,
                  /*reuse_a=*/false, /*reuse_b=*/false);
    }

    // --- store: lane (g, p<4) owns out[P0 + 8g + r][w=p] for r=0..7
    if (p < 4) {
        float* o = out + (P0 + 8 * g) * 4 + p;
        #pragma unroll
        for (int r = 0; r < 8; ++r) {
            o[r * 4] = acc[r];
        }
    }
}

// ---------------------------------------------------------------------------
extern "C" void kernel_launch(void* const* d_in, const int* in_sizes, int n_in,
                              void* d_out, int out_size, void* d_ws, size_t ws_size,
                              hipStream_t stream) {
    const float* img = (const float*)d_in[0];   // [16][192][192] f32
    const float* wts = (const float*)d_in[1];   // [2][4] f32
    float* out = (float*)d_out;                 // [16][191][191][4] f32
    float* M84 = (float*)d_ws;                  // needs 84*16*4 = 5376 B scratch

    quonv_setup_kernel<<<1, 512, 0, stream>>>(wts, M84);

    const int threads = 256;                            // 8 waves / block
    const int blocks  = (N_TILES + 7) / 8;              // one tile per wave
    quonv_main_kernel<<<blocks, threads, 0, stream>>>(img, M84, out);
}